// FeatureResidual_7636451852614
// MI455X (gfx1250) — compile-verified
//
#include <hip/hip_runtime.h>

typedef __attribute__((ext_vector_type(2))) float v2f;
typedef __attribute__((ext_vector_type(4))) float v4f;
typedef __attribute__((ext_vector_type(8))) float v8f;

#define N_ROWS     100000
#define N_COLS     40
#define K_DIM      8
#define F_DIM      32
#define BATCH      1024
#define ROW_TILES  (N_ROWS / 16)    // 6250
#define BATCH_TILES (BATCH / 16)    // 64
#define ROW_SPLIT  16
#define WAVES_PER_BLOCK 8
#define WAVES_PER_BT (ROW_SPLIT * WAVES_PER_BLOCK)  // 128 waves cooperate per batch tile

// Monotonic float->uint mapping so unsigned compare == float compare.
__device__ __forceinline__ unsigned int ordered_f32(float f) {
    unsigned int b = __float_as_uint(f);
    return b ^ (((unsigned int)((int)b >> 31)) | 0x80000000u);
}

__global__ __launch_bounds__(256) void init_ws_kernel(unsigned long long* __restrict__ ws) {
    int i = blockIdx.x * 256 + threadIdx.x;
    if (i < BATCH) ws[i] = ~0ull;
}

// One wave computes 16(batch) x 16(rows) score tiles via V_WMMA_F32_16X16X4_F32 (K=8 -> 2 WMMAs).
__global__ __launch_bounds__(256) void argmin_kernel(
    const float* __restrict__ qk,   // (BATCH, K_DIM)
    const float* __restrict__ tbl,  // (N_ROWS, N_COLS), keys in cols [0,8)
    unsigned long long* __restrict__ ws) {

    const int  lane = threadIdx.x & 31;
    const int  wave = threadIdx.x >> 5;
    const int  bt    = blockIdx.x / ROW_SPLIT;  // batch tile [0,64)
    const int  split = blockIdx.x % ROW_SPLIT;  // row split  [0,16)
    const bool hi  = (lane & 16) != 0;          // lanes 16-31 supply hardware K slots 2,3
    const int  l16 = lane & 15;

    // ---- A matrix (queries), loaded once per wave as two b128.
    // f32 16x4 A layout: lanes 0-15 = M, VGPR0/1 hold K=0/1 (lanes<16) or K=2/3 (lanes>=16).
    const v4f qv0 = *(const v4f*)(qk + (bt * 16 + l16) * K_DIM);      // q[m][0..3]
    const v4f qv1 = *(const v4f*)(qk + (bt * 16 + l16) * K_DIM + 4);  // q[m][4..7]
    v2f a0, a1;
    a0.x = hi ? qv0.z : qv0.x;  a0.y = hi ? qv0.w : qv0.y;   // K-chunk 0: k=0..3
    a1.x = hi ? qv1.z : qv1.x;  a1.y = hi ? qv1.w : qv1.y;   // K-chunk 1: k=4..7

    // Running argmin per accumulator (strict '<' + monotonically increasing r per lane
    // => lowest index wins ties, matching jnp.argmin).
    float        best_val[8];
    unsigned int best_idx[8];
#pragma unroll
    for (int v = 0; v < 8; ++v) { best_val[v] = __int_as_float(0x7F800000); best_idx[v] = 0u; }

    // Wave-uniform loop bound, forced scalar so EXEC stays all-ones (WMMA requirement)
    // and loop control is pure SALU.
    const int wave_id = __builtin_amdgcn_readfirstlane(split * WAVES_PER_BLOCK + wave);

    for (int t = wave_id; t < ROW_TILES; t += WAVES_PER_BT) {
        const int r = t * 16 + l16;             // this lane's row (column N of the tile)
        const float* krow = tbl + (long)r * N_COLS;

        // Whole 8-float key row per lane: exactly two b128 loads, hi-independent addresses.
        const v4f k0 = *(const v4f*)(krow);      // k[0..3]
        const v4f k1 = *(const v4f*)(krow + 4);  // k[4..7]

        // Prefetch next tile's row into the NEAR caches. No scope modifier => SCOPE 0 (WGP):
        // per ISA this is the only scope that pulls into all cache levels; SE/DEV/SYS scopes
        // (all __builtin_prefetch can produce) only warm GL2, which is already hot.
        {
            int rn = r + WAVES_PER_BT * 16;
            rn = rn < N_ROWS ? rn : (N_ROWS - 1);
            const float* pnext = tbl + (long)rn * N_COLS;
            asm volatile("global_prefetch_b8 %0, off" : : "v"(pnext));
        }

        // B matrix 4x16: lanes hold N=l16; this lane's two hardware-K values selected
        // via loop-invariant mask (4 cndmasks).
        v2f b0, b1;
        b0.x = hi ? k0.z : k0.x;  b0.y = hi ? k0.w : k0.y;
        b1.x = hi ? k1.z : k1.x;  b1.y = hi ? k1.w : k1.y;

        // k_sq[r] directly from the full row vectors (pk_mul/pk_add friendly).
        float ksq = (k0.x * k0.x + k0.y * k0.y) + (k0.z * k0.z + k0.w * k0.w)
                  + (k1.x * k1.x + k1.y * k1.y) + (k1.z * k1.z + k1.w * k1.w);

        v8f c = {};
        c = __builtin_amdgcn_wmma_f32_16x16x4_f32(false, a0, false, b0, (short)0, c, false, false);
        c = __builtin_amdgcn_wmma_f32_16x16x4_f32(false, a1, false, b1, (short)0, c, false, false);

        // C/D layout: VGPR v -> M = v + 8*hi, N = l16. Same row r for all 8 accumulators.
#pragma unroll
        for (int v = 0; v < 8; ++v) {
            float score = __builtin_fmaf(-2.0f, c[v], ksq);   // q_sq constant per batch: omitted
            bool lt = score < best_val[v];
            best_val[v] = lt ? score : best_val[v];
            best_idx[v] = lt ? (unsigned int)r : best_idx[v];
        }
    }

    // Pack (orderable value, index) once; u64 min == lexicographic (value, lowest index).
    unsigned long long best[8];
#pragma unroll
    for (int v = 0; v < 8; ++v)
        best[v] = ((unsigned long long)ordered_f32(best_val[v]) << 32) | best_idx[v];

    // Reduce across the 16 lanes of each half (masks < 16 never cross halves).
#pragma unroll
    for (int mask = 8; mask >= 1; mask >>= 1) {
#pragma unroll
        for (int v = 0; v < 8; ++v) {
            unsigned long long o = __shfl_xor(best[v], mask, 32);
            best[v] = (o < best[v]) ? o : best[v];
        }
    }

    if (l16 == 0) {
        const int mbase = bt * 16 + (hi ? 8 : 0);
#pragma unroll
        for (int v = 0; v < 8; ++v)
            atomicMin(ws + mbase + v, best[v]);  // global_atomic_min_u64
    }
}

__global__ __launch_bounds__(256) void gather_kernel(
    const float* __restrict__ feats,            // (BATCH, F_DIM)
    const float* __restrict__ tbl,              // (N_ROWS, N_COLS), features in cols [8,40)
    const unsigned long long* __restrict__ ws,  // packed (score, idx) per batch
    float* __restrict__ out) {
    int e = blockIdx.x * 256 + threadIdx.x;
    if (e >= BATCH * F_DIM) return;
    int b = e >> 5;
    int j = e & 31;
    unsigned int idx = (unsigned int)(ws[b] & 0xFFFFFFFFull);
    out[e] = feats[e] - tbl[(long)idx * N_COLS + K_DIM + j];
}

extern "C" void kernel_launch(void* const* d_in, const int* in_sizes, int n_in,
                              void* d_out, int out_size, void* d_ws, size_t ws_size,
                              hipStream_t stream) {
    const float* qk    = (const float*)d_in[0];  // predicted_key (1024, 8)
    const float* feats = (const float*)d_in[1];  // features      (1024, 32)
    const float* tbl   = (const float*)d_in[2];  // lookup_table  (100000, 40)
    // d_in[3]/d_in[4] are arange index vectors per the reference; layouts hardcoded.
    unsigned long long* ws = (unsigned long long*)d_ws;  // 1024 x u64 = 8 KB
    float* out = (float*)d_out;

    init_ws_kernel<<<(BATCH + 255) / 256, 256, 0, stream>>>(ws);
    argmin_kernel<<<BATCH_TILES * ROW_SPLIT, 256, 0, stream>>>(qk, tbl, ws);
    gather_kernel<<<(BATCH * F_DIM + 255) / 256, 256, 0, stream>>>(feats, tbl, ws, out);
}